// RoiPooling2D_44873818309085
// MI455X (gfx1250) — compile-verified
//
#include <hip/hip_runtime.h>
#include <hip/hip_bf16.h>
#include <stdint.h>

#define POOL   7
#define NROIS  512
#define IMG_H  64
#define IMG_W  64
#define CH     1024   // channels, innermost & contiguous (NHWC)

// ---------------- CDNA5 async global->LDS (16B per lane, ASYNCcnt) ----------

typedef int v4i_b128 __attribute__((vector_size(16)));
typedef __attribute__((address_space(1))) v4i_b128* g_b128_p;   // global
typedef __attribute__((address_space(3))) v4i_b128* l_b128_p;   // LDS

__device__ __forceinline__ void async_copy_b128(const float* gsrc, float* ldst) {
#if __has_builtin(__builtin_amdgcn_global_load_async_to_lds_b128)
    __builtin_amdgcn_global_load_async_to_lds_b128(
        (g_b128_p)(uintptr_t)gsrc,
        (l_b128_p)(uint32_t)(uintptr_t)ldst,
        0, 0);
#else
    // GV mode: vdst = LDS byte offset, vaddr = 64-bit global address
    unsigned loff = (unsigned)(uintptr_t)ldst;
    asm volatile("global_load_async_to_lds_b128 %0, %1, off"
                 :: "v"(loff), "v"(gsrc) : "memory");
#endif
}

#if __has_builtin(__builtin_amdgcn_s_wait_asynccnt)
#define WAIT_ASYNC(n) do { __builtin_amdgcn_s_wait_asynccnt(n); \
                           asm volatile("" ::: "memory"); } while (0)
#else
#define WAIT_ASYNC(n) asm volatile("s_wait_asynccnt " #n ::: "memory")
#endif

// ---------------------------------------------------------------------------

__device__ __forceinline__ float lerp2d(float a00, float a01, float a10,
                                        float a11, float fx, float fy) {
    float top = a00 + (a01 - a00) * fx;
    float bot = a10 + (a11 - a10) * fx;
    return top + (bot - top) * fy;
}

// One block per (roi, py): pipelines the 7 px cells through a double-buffered
// LDS stage (2 cells x 4 rows x 1024 f32 = 32 KB).
__global__ __launch_bounds__(256)
void roi_pool_async_kernel(const float* __restrict__ img,
                           const int*   __restrict__ rois,
                           float*       __restrict__ out) {
    __shared__ __align__(16) float lds[2 * 4 * CH];   // 32 KB

    const int blk = blockIdx.x;          // 0 .. NROIS*POOL-1
    const int r   = blk / POOL;
    const int py  = blk - r * POOL;
    const int t   = threadIdx.x;         // channel group (float4)
    const int co  = 4 * t;               // channel offset of this thread

    const int4 roi = ((const int4*)rois)[r];
    const int rx = roi.x, ry = roi.y, rw = roi.z, rh = roi.w;

    const float hs = (float)rh / (float)POOL;
    const float ws = (float)rw / (float)POOL;

    // y interpolation parameters (shared by all 7 cells of this block)
    const float sy = (float)py * hs;
    const int   y0 = (int)sy;                 // sy >= 0 -> trunc == floor
    const float fy = sy - (float)y0;
    const int   y1 = (y0 + 1 < rh) ? (y0 + 1) : (rh - 1);
    const long  row0 = (long)(ry + y0) * IMG_W;
    const long  row1 = (long)(ry + y1) * IMG_W;

    auto cellx = [&](int px, int& gx0, int& gx1, float& fx) {
        const float sx = (float)px * ws;
        const int   x0 = (int)sx;
        fx = sx - (float)x0;
        const int x1 = (x0 + 1 < rw) ? (x0 + 1) : (rw - 1);
        gx0 = rx + x0;
        gx1 = rx + x1;
    };

    // Issue the 4 gathered pixel rows of one cell into LDS buffer `buf`.
    auto issue = [&](int buf, int gx0, int gx1) {
        float* b = &lds[buf * (4 * CH)];
        async_copy_b128(img + (row0 + gx0) * CH + co, b + 0 * CH + co);
        async_copy_b128(img + (row0 + gx1) * CH + co, b + 1 * CH + co);
        async_copy_b128(img + (row1 + gx0) * CH + co, b + 2 * CH + co);
        async_copy_b128(img + (row1 + gx1) * CH + co, b + 3 * CH + co);
    };

    // Prologue: start cell 0.
    int gx0, gx1; float fx;
    cellx(0, gx0, gx1, fx);
    issue(0, gx0, gx1);

    // out layout: (roi, py, px, c) -> ((blk*POOL + px)*CH + c), float4 units
    float4* outv = (float4*)out + (size_t)blk * POOL * (CH / 4) + t;

    for (int px = 0; px < POOL; ++px) {
        int ngx0 = 0, ngx1 = 0; float nfx = 0.0f;
        if (px + 1 < POOL) {
            cellx(px + 1, ngx0, ngx1, nfx);
            issue((px + 1) & 1, ngx0, ngx1);   // prefetch next cell
            WAIT_ASYNC(4);                     // only next cell's 4 in flight
        } else {
            WAIT_ASYNC(0);                     // drain for the last cell
        }

        const float* b = &lds[(px & 1) * (4 * CH)];
        float4 v00 = *(const float4*)(b + 0 * CH + co);
        float4 v01 = *(const float4*)(b + 1 * CH + co);
        float4 v10 = *(const float4*)(b + 2 * CH + co);
        float4 v11 = *(const float4*)(b + 3 * CH + co);

        float4 res;
        res.x = lerp2d(v00.x, v01.x, v10.x, v11.x, fx, fy);
        res.y = lerp2d(v00.y, v01.y, v10.y, v11.y, fx, fy);
        res.z = lerp2d(v00.z, v01.z, v10.z, v11.z, fx, fy);
        res.w = lerp2d(v00.w, v01.w, v10.w, v11.w, fx, fy);

        outv[px * (CH / 4)] = res;

        gx0 = ngx0; gx1 = ngx1; fx = nfx;
    }
}

extern "C" void kernel_launch(void* const* d_in, const int* in_sizes, int n_in,
                              void* d_out, int out_size, void* d_ws, size_t ws_size,
                              hipStream_t stream) {
    const float* img  = (const float*)d_in[0];   // (1, 64, 64, 1024) f32
    const int*   rois = (const int*)d_in[1];     // (1, 512, 4) i32
    float*       out  = (float*)d_out;           // (1, 512, 7, 7, 1024) f32

    dim3 grid(NROIS * POOL);   // 3584 blocks: one per (roi, py)
    dim3 block(256);           // 8 wave32s; thread t owns channels 4t..4t+3
    hipLaunchKernelGGL(roi_pool_async_kernel, grid, block, 0, stream,
                       img, rois, out);
}